// SimpleGraphSAGE_53197464928385
// MI455X (gfx1250) — compile-verified
//
#include <hip/hip_runtime.h>
#include <hip/hip_bf16.h>

// ---------------------------------------------------------------------------
// GraphSAGE (3x SAGEConv mean-aggr + relu, hstack output) for MI455X gfx1250.
// Aggregation is bandwidth/atomic bound (L2-resident); GEMMs use the fp32
// matrix pipe V_WMMA_F32_16X16X4_F32 (exact fp32, plenty fast for 6.5 GFLOP).
// ---------------------------------------------------------------------------

typedef __attribute__((ext_vector_type(2))) float v2f;
typedef __attribute__((ext_vector_type(8))) float v8f;

#define SAGE_N_NODES 50000
#define SAGE_N_EDGES 800000

// ---------------- simple fills ----------------
__global__ __launch_bounds__(256) void sage_fill0(float* __restrict__ p, long n) {
    long t = (long)blockIdx.x * blockDim.x + threadIdx.x;
    if (t < n) p[t] = 0.0f;
}

// degree count: cnt[dst[e]] += 1
__global__ __launch_bounds__(256) void sage_degree(const int* __restrict__ dst,
                                                   float* __restrict__ cnt, int E) {
    int e = blockIdx.x * blockDim.x + threadIdx.x;
    if (e < E) atomicAdd(cnt + dst[e], 1.0f);
}

// inv = 1 / max(cnt, 1)
__global__ __launch_bounds__(256) void sage_recip(float* __restrict__ c, int n) {
    int t = blockIdx.x * blockDim.x + threadIdx.x;
    if (t < n) c[t] = 1.0f / fmaxf(c[t], 1.0f);
}

// ---------------- edge scatter-add (mean numerator) ----------------
// one thread per (edge, 4-channel chunk): float4 gather + 4 f32 atomics.
__global__ __launch_bounds__(256) void sage_scatter(const int* __restrict__ src,
                                                    const int* __restrict__ dst,
                                                    const float* __restrict__ feat,
                                                    int sf,            // feature row stride
                                                    float* __restrict__ agg,
                                                    int d)             // channels (agg stride)
{
    int chunks = d >> 2;
    long t = (long)blockIdx.x * blockDim.x + threadIdx.x;
    if (t >= (long)SAGE_N_EDGES * chunks) return;
    long e  = t / chunks;
    int  c4 = (int)(t % chunks) << 2;
    int  s  = src[e];
    int  dd = dst[e];
    const float4 v = *(const float4*)(feat + (long)s * sf + c4);
    float* ap = agg + (long)dd * d + c4;
    atomicAdd(ap + 0, v.x);
    atomicAdd(ap + 1, v.y);
    atomicAdd(ap + 2, v.z);
    atomicAdd(ap + 3, v.w);
}

// ---------------- fused dual-GEMM + bias + relu via fp32 WMMA ----------------
// out[m, n] = relu( sum_k agg[m,k]*inv[m] * Wl[n,k] + sum_k xin[m,k] * Wr[n,k] + b[n] )
// One wave32 computes one 16x16 output tile; K-step = 4 (V_WMMA_F32_16X16X4_F32).
__global__ __launch_bounds__(256) void sage_gemm_relu(
    const float* __restrict__ agg,   // [N, K] contiguous
    const float* __restrict__ inv,   // [N]
    const float* __restrict__ xin,   // self features base (row stride sx)
    int sx,
    const float* __restrict__ Wl,    // [dout, K] row-major
    const float* __restrict__ Wr,    // [dout, K] row-major
    const float* __restrict__ bias,  // [dout]
    float* __restrict__ out,         // base already offset to this layer's column slice
    int so,                          // out row stride (320)
    int N, int K, int dout)
{
    const int wave = threadIdx.x >> 5;   // 8 waves / block (wave32)
    const int lane = threadIdx.x & 31;
    const int tiles_n = dout >> 4;
    const int tiles_m = (N + 15) >> 4;
    const long tile = (long)blockIdx.x * 8 + wave;
    if (tile >= (long)tiles_m * tiles_n) return;          // wave-uniform: EXEC stays all-1s
    const int tile_m = (int)(tile / tiles_n);
    const int tile_n = (int)(tile % tiles_n);

    // A fragment (16x4 f32): lane L -> row M = L&15, K-pair base = 2*(L>=16)
    const int m   = tile_m * 16 + (lane & 15);
    const int kb  = (lane >> 4) << 1;                      // 0 or 2
    // B fragment (4x16 f32): lane L -> col N = L&15, same K-pair striping
    const int n0  = tile_n * 16 + (lane & 15);

    const float invm = inv[m];
    const float* aggp = agg + (long)m * K + kb;
    const float* xp   = xin + (long)m * sx + kb;
    const float* wlp  = Wl  + (long)n0 * K + kb;
    const float* wrp  = Wr  + (long)n0 * K + kb;

    v8f acc = {};
#pragma unroll 4
    for (int k = 0; k < K; k += 4) {
        v2f a1 = *(const v2f*)(aggp + k);
        a1 *= invm;                                        // fuse mean scaling
        v2f b1 = *(const v2f*)(wlp + k);
        acc = __builtin_amdgcn_wmma_f32_16x16x4_f32(
            false, a1, false, b1, (short)0, acc, false, false);
        v2f a2 = *(const v2f*)(xp + k);
        v2f b2 = *(const v2f*)(wrp + k);
        acc = __builtin_amdgcn_wmma_f32_16x16x4_f32(
            false, a2, false, b2, (short)0, acc, false, false);
    }

    // C/D layout: lane column = n0; VGPR p holds row p + 8*(lane>=16)
    const float bv = bias[n0];
    const int rbase = tile_m * 16 + ((lane >> 4) << 3);
#pragma unroll
    for (int p = 0; p < 8; ++p) {
        float v = acc[p] + bv;
        out[(long)(rbase + p) * so + n0] = v > 0.0f ? v : 0.0f;
    }
}

// ---------------------------------------------------------------------------
extern "C" void kernel_launch(void* const* d_in, const int* in_sizes, int n_in,
                              void* d_out, int out_size, void* d_ws, size_t ws_size,
                              hipStream_t stream) {
    const float* x    = (const float*)d_in[0];
    const int*   eidx = (const int*)d_in[1];           // [2, E] flat
    const int*   src  = eidx;
    const int*   dst  = eidx + SAGE_N_EDGES;
    const float* W1l  = (const float*)d_in[2];
    const float* b1   = (const float*)d_in[3];
    const float* W1r  = (const float*)d_in[4];
    const float* W2l  = (const float*)d_in[5];
    const float* b2   = (const float*)d_in[6];
    const float* W2r  = (const float*)d_in[7];
    const float* W3l  = (const float*)d_in[8];
    const float* b3   = (const float*)d_in[9];
    const float* W3r  = (const float*)d_in[10];

    float* out = (float*)d_out;                        // [N, 320] hstack(h1,h2,h3)
    const int SO = 320;

    float* agg = (float*)d_ws;                         // up to N*128 f32 = 25.6 MB
    float* inv = agg + (long)SAGE_N_NODES * 128;       // N f32

    const int T = 256;
    auto blocks = [](long n, int t) { return (int)((n + t - 1) / t); };

    // ---- degree (shared by all layers) ----
    sage_fill0 <<<blocks(SAGE_N_NODES, T), T, 0, stream>>>(inv, SAGE_N_NODES);
    sage_degree<<<blocks(SAGE_N_EDGES, T), T, 0, stream>>>(dst, inv, SAGE_N_EDGES);
    sage_recip <<<blocks(SAGE_N_NODES, T), T, 0, stream>>>(inv, SAGE_N_NODES);

    // ---- layer 1: 128 -> 128, input x (stride 128), out cols [0,128) ----
    {
        const int K = 128, D = 128;
        sage_fill0  <<<blocks((long)SAGE_N_NODES * K, T), T, 0, stream>>>(agg, (long)SAGE_N_NODES * K);
        sage_scatter<<<blocks((long)SAGE_N_EDGES * (K >> 2), T), T, 0, stream>>>(src, dst, x, K, agg, K);
        long tiles = (long)((SAGE_N_NODES + 15) / 16) * (D / 16);
        sage_gemm_relu<<<blocks(tiles, 8), T, 0, stream>>>(agg, inv, x, K, W1l, W1r, b1,
                                                           out + 0, SO, SAGE_N_NODES, K, D);
    }
    // ---- layer 2: 128 -> 64, input h1 = out[:,0:128) (stride 320), out cols [128,192) ----
    {
        const int K = 128, D = 64;
        sage_fill0  <<<blocks((long)SAGE_N_NODES * K, T), T, 0, stream>>>(agg, (long)SAGE_N_NODES * K);
        sage_scatter<<<blocks((long)SAGE_N_EDGES * (K >> 2), T), T, 0, stream>>>(src, dst, out + 0, SO, agg, K);
        long tiles = (long)((SAGE_N_NODES + 15) / 16) * (D / 16);
        sage_gemm_relu<<<blocks(tiles, 8), T, 0, stream>>>(agg, inv, out + 0, SO, W2l, W2r, b2,
                                                           out + 128, SO, SAGE_N_NODES, K, D);
    }
    // ---- layer 3: 64 -> 128, input h2 = out[:,128:192) (stride 320), out cols [192,320) ----
    {
        const int K = 64, D = 128;
        sage_fill0  <<<blocks((long)SAGE_N_NODES * K, T), T, 0, stream>>>(agg, (long)SAGE_N_NODES * K);
        sage_scatter<<<blocks((long)SAGE_N_EDGES * (K >> 2), T), T, 0, stream>>>(src, dst, out + 128, SO, agg, K);
        long tiles = (long)((SAGE_N_NODES + 15) / 16) * (D / 16);
        sage_gemm_relu<<<blocks(tiles, 8), T, 0, stream>>>(agg, inv, out + 128, SO, W3l, W3r, b3,
                                                           out + 192, SO, SAGE_N_NODES, K, D);
    }
}